// BiTemperedWeightedLoss_661424963928
// MI455X (gfx1250) — compile-verified
//
#include <hip/hip_runtime.h>
#include <cmath>

typedef unsigned int u32;
typedef __attribute__((ext_vector_type(4))) u32 v4u;
typedef __attribute__((ext_vector_type(8))) int  v8i;
typedef __attribute__((ext_vector_type(4))) int  v4i;

#define WG_ROWS 8
#define MAXC    1024
#define NITER   5

__device__ __forceinline__ float fast_rcp(float x) {
#if __has_builtin(__builtin_amdgcn_rcpf)
  return __builtin_amdgcn_rcpf(x);
#else
  return 1.0f / x;
#endif
}

__device__ __forceinline__ float wave_sum(float v) {
#pragma unroll
  for (int d = 16; d > 0; d >>= 1) v += __shfl_xor(v, d, 32);
  return v;
}

__device__ __forceinline__ float wave_max(float v) {
#pragma unroll
  for (int d = 16; d > 0; d >>= 1) v = fmaxf(v, __shfl_xor(v, d, 32));
  return v;
}

// ---------------------------------------------------------------------------
// Kernel 1: sum(weight) -> ws[0]
// ---------------------------------------------------------------------------
__global__ __launch_bounds__(256) void btl_wsum_kernel(const float* __restrict__ w,
                                                       int C, float* __restrict__ out) {
  __shared__ float sp[8];
  float acc = 0.f;
  for (int i = threadIdx.x; i < C; i += 256) acc += w[i];
  acc = wave_sum(acc);
  if ((threadIdx.x & 31) == 0) sp[threadIdx.x >> 5] = acc;
  __syncthreads();
  if (threadIdx.x == 0) {
    float t = 0.f;
#pragma unroll
    for (int i = 0; i < 8; ++i) t += sp[i];
    out[0] = t;
  }
}

// ---------------------------------------------------------------------------
// Kernel 2: main. 8 waves per block, one row (C<=1024 fp32) per wave.
// TDM stages the 8xC logit tile + weight vector into LDS; each wave then
// keeps its row register-resident across all 6 normalization passes.
// ---------------------------------------------------------------------------
__global__ __launch_bounds__(256) void btl_main_kernel(
    const float* __restrict__ logits, const int* __restrict__ truth,
    const float* __restrict__ weight, const float* __restrict__ wsumP,
    float* __restrict__ partials, int B, int C,
    float Kon, float Koff, float on5, float off5) {
  __shared__ float ldsL[WG_ROWS * MAXC];
  __shared__ float ldsW[MAXC];
  __shared__ float s_part[WG_ROWS];

  const int lane = threadIdx.x & 31;
  const int wv   = threadIdx.x >> 5;
  const int row0 = blockIdx.x * WG_ROWS;

#if __has_builtin(__builtin_amdgcn_tensor_load_to_lds) && __has_builtin(__builtin_amdgcn_s_wait_tensorcnt)
  if (threadIdx.x < 32) {  // TDM ops are per-wave; issue once from wave 0
    const u32 uc = (u32)C;
    // ---- D# for logits tile: 2D tensor (dim0=C, dim1=rows_remaining),
    //      tile C x WG_ROWS, 4-byte elements, contiguous into ldsL.
    unsigned long long ga = (unsigned long long)(uintptr_t)(logits + (size_t)row0 * (size_t)C);
    const u32 rows_rem = (u32)(B - row0);  // y-OOB rows read back as zero
    v4u g0;
    g0[0] = 1u;                                     // count=1 valid descriptor
    g0[1] = (u32)(uintptr_t)&ldsL[0];               // lds_addr (bytes)
    g0[2] = (u32)(ga & 0xffffffffu);                // global_addr[31:0]
    g0[3] = (u32)((ga >> 32) & 0x1ffffffu) | (2u << 30);  // addr[56:32] | type=2
    v8i g1;
    g1[0] = (int)(2u << 16);                              // wg_mask=0, data_size=4B
    g1[1] = (int)((uc & 0xffffu) << 16);                  // tensor_dim0[15:0]
    g1[2] = (int)((uc >> 16) | ((rows_rem & 0xffffu) << 16));   // dim0 hi | dim1 lo
    g1[3] = (int)((rows_rem >> 16) | ((uc & 0xffffu) << 16));   // dim1 hi | tile_dim0
    g1[4] = (int)WG_ROWS;                                 // tile_dim1 | tile_dim2=0
    g1[5] = (int)uc;                                      // tensor_dim0_stride lo
    g1[6] = 0;
    g1[7] = 0;
    v4i z4; z4[0] = 0; z4[1] = 0; z4[2] = 0; z4[3] = 0;
    v8i z8; z8[0] = 0; z8[1] = 0; z8[2] = 0; z8[3] = 0;
    z8[4] = 0; z8[5] = 0; z8[6] = 0; z8[7] = 0;
    __builtin_amdgcn_tensor_load_to_lds(g0, g1, z4, z4, z8, 0);

    // ---- D# for weight vector: 1 x C tile into ldsW.
    unsigned long long wa = (unsigned long long)(uintptr_t)weight;
    v4u h0;
    h0[0] = 1u;
    h0[1] = (u32)(uintptr_t)&ldsW[0];
    h0[2] = (u32)(wa & 0xffffffffu);
    h0[3] = (u32)((wa >> 32) & 0x1ffffffu) | (2u << 30);
    v8i h1;
    h1[0] = (int)(2u << 16);
    h1[1] = (int)((uc & 0xffffu) << 16);
    h1[2] = (int)((uc >> 16) | (1u << 16));   // tensor_dim1 = 1
    h1[3] = (int)((uc & 0xffffu) << 16);      // tile_dim0 = C
    h1[4] = 1;                                // tile_dim1 = 1
    h1[5] = (int)uc;
    h1[6] = 0;
    h1[7] = 0;
    __builtin_amdgcn_tensor_load_to_lds(h0, h1, z4, z4, z8, 0);

    __builtin_amdgcn_s_wait_tensorcnt(0);
  }
  __syncthreads();
#else
  for (int i = threadIdx.x; i < WG_ROWS * C; i += 256) {
    const int r = row0 + i / C;
    ldsL[i] = (r < B) ? logits[(size_t)row0 * (size_t)C + i] : 0.f;
  }
  for (int i = threadIdx.x; i < C; i += 256) ldsW[i] = weight[i];
  __syncthreads();
#endif

  const int row = row0 + wv;
  float loss = 0.f;
  if (row < B) {
    const float* rp = &ldsL[wv * C];
    float a0[8][4];
    float mymax = -3.402823466e38f;
#pragma unroll
    for (int kk = 0; kk < 8; ++kk) {
      const int o = kk * 128 + (lane << 2);
      if (o < C) {
        const float4 v = *reinterpret_cast<const float4*>(rp + o);
        a0[kk][0] = v.x; a0[kk][1] = v.y; a0[kk][2] = v.z; a0[kk][3] = v.w;
        mymax = fmaxf(mymax, fmaxf(fmaxf(v.x, v.y), fmaxf(v.z, v.w)));
      }
    }
    mymax = wave_max(mymax);
#pragma unroll
    for (int kk = 0; kk < 8; ++kk) {
      const int o = kk * 128 + (lane << 2);
      if (o < C) {
#pragma unroll
        for (int j = 0; j < 4; ++j) a0[kk][j] -= mymax;  // a0 <= 0
      }
    }

    // Fixed point: Z = sum (1 - 0.2*s*a0)^-5 ; s = Z^-0.2 (5 updates + final Z)
    float s = 1.0f, Z = 1.0f;
#pragma unroll
    for (int it = 0; it <= NITER; ++it) {
      const float c = -0.2f * s;
      float acc = 0.f;
#pragma unroll
      for (int kk = 0; kk < 8; ++kk) {
        const int o = kk * 128 + (lane << 2);
        if (o < C) {
#pragma unroll
          for (int j = 0; j < 4; ++j) {
            const float w  = fmaf(c, a0[kk][j], 1.0f);   // >= 1, relu is no-op
            const float w2 = w * w;
            const float w4 = w2 * w2;
            acc += fast_rcp(w4 * w);                     // (1-0.2a)^-5
          }
        }
      }
      Z = wave_sum(acc);
      if (it < NITER) s = exp2f(-0.2f * log2f(Z));       // Z^(1-t)
    }

    // nc - mu = -log_t(1/Z, 1.2) = 5*(Z^0.2 - 1)
    const float ncmm = 5.0f * (exp2f(0.2f * log2f(Z)) - 1.0f);
    const float cw   = fmaf(0.2f, ncmm, 1.0f);           // w = 1 - 0.2*a0 + 0.2*ncmm
    const int   tr    = truth[row];
    const float scale = (float)C / wsumP[0];             // pos_weight factor
    const float c12   = 1.0f / 1.2f;
    float acc = 0.f;
#pragma unroll
    for (int kk = 0; kk < 8; ++kk) {
      const int o = kk * 128 + (lane << 2);
      if (o < C) {
        const float4 pv = *reinterpret_cast<const float4*>(&ldsW[o]);
        const float pw[4] = {pv.x, pv.y, pv.z, pv.w};
#pragma unroll
        for (int j = 0; j < 4; ++j) {
          // p = w^-5  =>  p^0.2 = 1/w exactly, p^1.2 = w^-6 exactly.
          const float w  = fmaf(-0.2f, a0[kk][j], cw);
          const float r  = fast_rcp(w);
          const float r2 = r * r;
          const float r6 = r2 * r2 * r2;
          const bool  isT = (o + j) == tr;
          const float K   = isT ? Kon : Koff;    // oh*log_t(oh+eps) - oh^1.2/1.2
          const float oh5 = isT ? on5 : off5;    // 5*oh
          // loss_i = K + 5*oh*(1-r) + r^6/1.2, weighted by pos_weight
          const float lv = fmaf(r6, c12, fmaf(oh5, 1.0f - r, K));
          acc = fmaf(lv, pw[j] * scale, acc);
        }
      }
    }
    loss = wave_sum(acc);
  }

  if (lane == 0) s_part[wv] = loss;
  __syncthreads();
  if (threadIdx.x == 0) {
    float t = 0.f;
#pragma unroll
    for (int i = 0; i < WG_ROWS; ++i) t += s_part[i];
    partials[blockIdx.x] = t;
  }
}

// ---------------------------------------------------------------------------
// Kernel 3: deterministic reduction of per-block partials -> mean loss
// ---------------------------------------------------------------------------
__global__ __launch_bounds__(256) void btl_final_kernel(const float* __restrict__ partials,
                                                        int n, int B, float* __restrict__ out) {
  __shared__ float sp[8];
  float acc = 0.f;
  for (int i = threadIdx.x; i < n; i += 256) acc += partials[i];
  acc = wave_sum(acc);
  if ((threadIdx.x & 31) == 0) sp[threadIdx.x >> 5] = acc;
  __syncthreads();
  if (threadIdx.x == 0) {
    float t = 0.f;
#pragma unroll
    for (int i = 0; i < 8; ++i) t += sp[i];
    out[0] = t / (float)B;
  }
}

// ---------------------------------------------------------------------------
extern "C" void kernel_launch(void* const* d_in, const int* in_sizes, int n_in,
                              void* d_out, int out_size, void* d_ws, size_t ws_size,
                              hipStream_t stream) {
  const float* logits = (const float*)d_in[0];
  const int*   truth  = (const int*)d_in[1];
  const float* weight = (const float*)d_in[2];
  const int B = in_sizes[1];
  const int C = in_sizes[2];  // C=1000 <= MAXC, multiple of 4

  float* ws       = (float*)d_ws;
  float* wsum     = ws;        // ws[0]
  float* partials = ws + 64;   // 256B-aligned partial array, nBlocks floats
  const int nBlocks = (B + WG_ROWS - 1) / WG_ROWS;

  // Host-side one-hot/smoothing constants (double precision).
  const double Cn  = (double)C;
  const double S   = 0.05;                              // SMOOTHING
  const double off = S / (Cn - 1.0);
  const double on  = (1.0 - S * Cn / (Cn - 1.0)) + off;
  auto logt08 = [](double u) { return (std::pow(u, 0.2) - 1.0) / 0.2; };  // log_t(.,0.8)
  const double Kon  = on  * logt08(on  + 1e-10) - std::pow(on,  1.2) / 1.2;
  const double Koff = off * logt08(off + 1e-10) - std::pow(off, 1.2) / 1.2;

  btl_wsum_kernel<<<1, 256, 0, stream>>>(weight, C, wsum);
  btl_main_kernel<<<nBlocks, 256, 0, stream>>>(logits, truth, weight, wsum, partials,
                                               B, C, (float)Kon, (float)Koff,
                                               (float)(5.0 * on), (float)(5.0 * off));
  btl_final_kernel<<<1, 256, 0, stream>>>(partials, nBlocks, B, (float*)d_out);
}